// CTCLayer_68161130987914
// MI455X (gfx1250) — compile-verified
//
#include <hip/hip_runtime.h>
#include <stdint.h>

// CTC forward loss on MI455X (gfx1250, wave32).
// One wave32 per batch element. Lane l owns states s = 5l .. 5l+4 (S = 129).
// y_pred rows (512 floats = 2KB) are double-buffered in LDS via CDNA5 async
// global->LDS DMA (global_load_async_to_lds_b128, tracked by ASYNCcnt),
// overlapped with the log-sum-exp DP of the current step.

#define NEGF (-1e30f)
#define EPSF (1e-7f)

static constexpr int Tc = 512;
static constexpr int Cc = 512;
static constexpr int Lc = 64;
static constexpr int Sc = 2 * Lc + 1;   // 129

#define WAIT_ASYNC(n) asm volatile("s_wait_asynccnt " #n ::: "memory")

// Issue one full row (2048 B) as 4 x b128 per lane into LDS, asynchronously.
// INST_OFFSET is added to both the global and LDS addresses (ISA 15.18.3),
// so one base pair + three offsets covers the whole row.
__device__ __forceinline__ void issue_row_async(const float* gRow,
                                                uint32_t ldsBase, int lane) {
  uint64_t ga = (uint64_t)(uintptr_t)gRow + (uint32_t)(lane * 16);
  uint32_t la = ldsBase + (uint32_t)(lane * 16);
  asm volatile(
      "global_load_async_to_lds_b128 %0, %1, off\n\t"
      "global_load_async_to_lds_b128 %0, %1, off offset:512\n\t"
      "global_load_async_to_lds_b128 %0, %1, off offset:1024\n\t"
      "global_load_async_to_lds_b128 %0, %1, off offset:1536"
      :: "v"(la), "v"(ga) : "memory");
}

__global__ __launch_bounds__(32) void ctc_fwd_kernel(
    const int* __restrict__ y_true,     // [B, L]
    const float* __restrict__ y_pred,   // [B, T, C] softmax probs
    const int* __restrict__ in_len,     // [B]
    const int* __restrict__ lab_len,    // [B]
    float* __restrict__ out)            // [B]
{
  __shared__ __align__(16) float rowbuf[2][Cc];
  __shared__ float finals[2];

  const int b    = blockIdx.x;
  const int lane = threadIdx.x;
  const int tin  = in_len[b];
  const int lab  = lab_len[b];

  const uint32_t lds0 = (uint32_t)(uintptr_t)(&rowbuf[0][0]);
  const uint32_t lds1 = (uint32_t)(uintptr_t)(&rowbuf[1][0]);
  const float* yp_b = y_pred + (size_t)b * Tc * Cc;

  // Per-lane extended-label indices and skip-transition permission.
  int   ext[5];
  bool  allowb[5];
  bool  valid[5];
#pragma unroll
  for (int j = 0; j < 5; ++j) {
    int s = 5 * lane + j;
    bool v = (s < Sc);
    int e = Cc - 1;          // blank
    bool alw = false;
    if (v && (s & 1)) {
      int jj = (s - 1) >> 1;
      e = y_true[b * Lc + jj];
      if (s >= 2) {
        int pl = y_true[b * Lc + jj - 1];
        alw = (e != pl) && (e != Cc - 1);
      }
    }
    ext[j] = e;
    allowb[j] = alw;
    valid[j] = v;
  }

  // Prologue: DMA rows 0 (and 1) into the double buffer.
  issue_row_async(yp_b + 0 * Cc, lds0, lane);
  if (tin > 1) {
    issue_row_async(yp_b + 1 * Cc, lds1, lane);
    WAIT_ASYNC(4);           // row 0 resident (in-order completion per wave)
  } else {
    WAIT_ASYNC(0);
  }

  float a[5];
  float e_[5];

  // t = 0: alpha0
  {
    const volatile float* rb = &rowbuf[0][0];
#pragma unroll
    for (int j = 0; j < 5; ++j) {
      float p = rb[ext[j]];
      e_[j] = __logf(p + EPSF);
    }
#pragma unroll
    for (int j = 0; j < 5; ++j) {
      int s = 5 * lane + j;
      a[j] = (valid[j] && s < 2) ? e_[j] : NEGF;
    }
    if (2 < tin) {
      issue_row_async(yp_b + 2 * Cc, lds0, lane);
      WAIT_ASYNC(4);         // row 1 resident
    } else {
      WAIT_ASYNC(0);
    }
  }

  // Sequential DP over time.
  for (int t = 1; t < tin; ++t) {
    const int curb = t & 1;
    const volatile float* rb = &rowbuf[curb][0];
#pragma unroll
    for (int j = 0; j < 5; ++j) {
      float p = rb[ext[j]];
      e_[j] = __logf(p + EPSF);
    }

    // Cross-lane neighbors: state 5l-1 is a[4] of lane l-1, 5l-2 is a[3].
    float up4 = __shfl_up(a[4], 1);
    float up3 = __shfl_up(a[3], 1);
    if (lane == 0) { up4 = NEGF; up3 = NEGF; }

    float na[5];
#pragma unroll
    for (int j = 0; j < 5; ++j) {
      float p1 = (j == 0) ? up4 : a[j - 1];
      float p2 = (j == 0) ? up3 : ((j == 1) ? up4 : a[j - 2]);
      p2 = allowb[j] ? p2 : NEGF;
      float m = fmaxf(a[j], fmaxf(p1, p2));
      float ssum = __expf(a[j] - m) + __expf(p1 - m) + __expf(p2 - m);
      na[j] = m + __logf(ssum) + e_[j];
    }
#pragma unroll
    for (int j = 0; j < 5; ++j) a[j] = valid[j] ? na[j] : NEGF;

    // Prefetch row t+2 into the buffer we just finished reading.
    if (t + 2 < tin) {
      issue_row_async(yp_b + (size_t)(t + 2) * Cc, curb ? lds1 : lds0, lane);
      WAIT_ASYNC(4);         // row t+1 resident, row t+2 in flight
    } else {
      WAIT_ASYNC(0);
    }
  }

  // Final: loss = -logsumexp(alpha[2*lab], alpha[2*lab-1])
  int s1 = 2 * lab;
  int s2 = 2 * lab - 1;
  if (s1 > Sc - 1) s1 = Sc - 1;
  if (s1 < 0) s1 = 0;
  if (s2 > Sc - 1) s2 = Sc - 1;
  if (s2 < 0) s2 = 0;
#pragma unroll
  for (int j = 0; j < 5; ++j) {
    int s = 5 * lane + j;
    if (s == s1) finals[0] = a[j];
    if (s == s2) finals[1] = a[j];
  }
  __syncthreads();
  if (lane == 0) {
    volatile float* fin = finals;
    float f1 = fin[0], f2 = fin[1];
    float m = fmaxf(f1, f2);
    float l = m + __logf(__expf(f1 - m) + __expf(f2 - m));
    out[b] = -l;
  }
}

extern "C" void kernel_launch(void* const* d_in, const int* in_sizes, int n_in,
                              void* d_out, int out_size, void* d_ws, size_t ws_size,
                              hipStream_t stream) {
  const int*   y_true  = (const int*)d_in[0];
  const float* y_pred  = (const float*)d_in[1];
  const int*   in_len  = (const int*)d_in[2];
  const int*   lab_len = (const int*)d_in[3];
  float*       out     = (float*)d_out;
  const int Bn = in_sizes[2];   // input_length has B elements
  ctc_fwd_kernel<<<Bn, 32, 0, stream>>>(y_true, y_pred, in_len, lab_len, out);
}